// NNLoss_44513041056088
// MI455X (gfx1250) — compile-verified
//
#include <hip/hip_runtime.h>
#include <stdint.h>

#define PAD_VAL (-10000.0f)

static constexpr int Bn  = 16, Cn = 3, Hn = 256, Wn = 256;
static constexpr int TLX = 64, TLY = 32;         // output pixels per block
static constexpr int PX  = 8;                    // pixels per thread along x
static constexpr int NTX = TLX / PX;             // 8 threads in x
static constexpr int MAXR  = 3;                  // supports nh,nw <= 7
static constexpr int MAXTH = TLY + 2 * MAXR;     // 38
static constexpr int MAXTW = TLX + 2 * MAXR;     // 70

// CDNA5 async global->LDS copy path (ASYNCcnt). Guarded so the file compiles
// on toolchains that lack the builtin (falls back to load+ds_store).
#if defined(__HIP_DEVICE_COMPILE__) && __has_builtin(__builtin_amdgcn_global_load_async_to_lds_b32)
#define USE_ASYNC_LDS 1
#endif

typedef __attribute__((address_space(1))) int as1_int;   // global
typedef __attribute__((address_space(3))) int as3_int;   // LDS

__device__ __forceinline__ void nnloss_body(
    int nh, int nw,
    const float* __restrict__ pred, const float* __restrict__ gt,
    float* __restrict__ partials, float* tile, float* wsum)
{
    const int tileH = TLY + nh - 1;
    const int tileW = TLX + nw - 1;
    const int per   = tileH * tileW;

    const int tx  = threadIdx.x;               // 0..7
    const int ty  = threadIdx.y;               // 0..31
    const int tid = ty * NTX + tx;             // 0..255
    const int b   = blockIdx.z;
    // Reference pads H by nw//2 and W by nh//2.
    const int gy0 = (int)blockIdx.y * TLY - (nw >> 1);
    const int gx0 = (int)blockIdx.x * TLX - (nh >> 1);

    // ---- Stage gt tile (with halo) into LDS via async copies ----
    const float* gbase = gt + (size_t)b * Cn * Hn * Wn;
    const int total = Cn * per;
    for (int e = tid; e < total; e += 256) {
        const int c   = e / per;
        const int rem = e - c * per;
        const int r   = rem / tileW;
        const int col = rem - r * tileW;
        const int gy = gy0 + r;
        const int gx = gx0 + col;
        if ((gy >= 0) && (gy < Hn) && (gx >= 0) && (gx < Wn)) {
            const float* src = gbase + (size_t)c * Hn * Wn + (size_t)gy * Wn + gx;
#ifdef USE_ASYNC_LDS
            __builtin_amdgcn_global_load_async_to_lds_b32(
                (as1_int*)(uintptr_t)src,
                (as3_int*)(uint32_t)(uintptr_t)(&tile[e]),
                0, 0);
#else
            tile[e] = *src;
#endif
        } else {
            tile[e] = PAD_VAL;     // plain DS store, disjoint addresses
        }
    }
#ifdef USE_ASYNC_LDS
#if __has_builtin(__builtin_amdgcn_s_wait_asynccnt)
    __builtin_amdgcn_s_wait_asynccnt(0);
#else
    asm volatile("s_wait_asynccnt 0" ::: "memory");
#endif
#endif
    __syncthreads();

    // ---- Per-pixel min, 8 pixels per thread along x ----
    const int gy = (int)blockIdx.y * TLY + ty;
    const int x0 = (int)blockIdx.x * TLX + tx * PX;

    float pr[Cn][PX];
#pragma unroll
    for (int c = 0; c < Cn; ++c) {
        const float* pp = pred + ((size_t)(b * Cn + c) * Hn + gy) * Wn + x0;
#pragma unroll
        for (int p = 0; p < PX; ++p) pr[c][p] = pp[p];
    }

    float mn[PX];
#pragma unroll
    for (int p = 0; p < PX; ++p) mn[p] = 3.4e38f;

    if (nh == 5 && nw == 5) {
        // Specialized: 12-float window per (row, channel) via 3x ds_load_b128.
#pragma unroll
        for (int i = 0; i < 5; ++i) {
            float w[Cn][12];
#pragma unroll
            for (int c = 0; c < Cn; ++c) {
                const float4* wbase = reinterpret_cast<const float4*>(
                    &tile[(c * tileH + ty + i) * tileW + tx * PX]);
                const float4 a = wbase[0], bb = wbase[1], d4 = wbase[2];
                w[c][0] = a.x;  w[c][1] = a.y;  w[c][2]  = a.z;  w[c][3]  = a.w;
                w[c][4] = bb.x; w[c][5] = bb.y; w[c][6]  = bb.z; w[c][7]  = bb.w;
                w[c][8] = d4.x; w[c][9] = d4.y; w[c][10] = d4.z; w[c][11] = d4.w;
            }
#pragma unroll
            for (int j = 0; j < 5; ++j) {
#pragma unroll
                for (int p = 0; p < PX; ++p) {
                    const float d = fabsf(w[0][p + j] - pr[0][p])
                                  + fabsf(w[1][p + j] - pr[1][p])
                                  + fabsf(w[2][p + j] - pr[2][p]);
                    mn[p] = fminf(mn[p], d);
                }
            }
        }
    } else {
        // Generic fallback (scalar LDS reads).
        for (int i = 0; i < nh; ++i) {
            const float* r0 = &tile[(ty + i) * tileW + tx * PX];
            const float* r1 = r0 + per;
            const float* r2 = r1 + per;
            for (int j = 0; j < nw; ++j) {
#pragma unroll
                for (int p = 0; p < PX; ++p) {
                    const float d = fabsf(r0[p + j] - pr[0][p])
                                  + fabsf(r1[p + j] - pr[1][p])
                                  + fabsf(r2[p + j] - pr[2][p]);
                    mn[p] = fminf(mn[p], d);
                }
            }
        }
    }

    float s = 0.0f;
#pragma unroll
    for (int p = 0; p < PX; ++p) s += mn[p];

    // ---- Deterministic block reduction (wave32 shuffle + LDS) ----
#pragma unroll
    for (int off = 16; off > 0; off >>= 1)
        s += __shfl_xor(s, off, 32);
    if ((tid & 31) == 0) wsum[tid >> 5] = s;
    __syncthreads();
    if (tid == 0) {
        float t = 0.0f;
#pragma unroll
        for (int wv = 0; wv < 8; ++wv) t += wsum[wv];
        partials[((size_t)blockIdx.z * gridDim.y + blockIdx.y) * gridDim.x
                 + blockIdx.x] = t;
    }
}

__global__ __launch_bounds__(256) void nnloss_min_tile(
    const float* __restrict__ pred, const float* __restrict__ gt,
    const int* __restrict__ nh_p, const int* __restrict__ nw_p,
    float* __restrict__ partials)
{
    __shared__ __align__(16) float tile[Cn * MAXTH * MAXTW];
    __shared__ float wsum[8];

    const int nh = nh_p[0];
    const int nw = nw_p[0];
    if (nh == 5 && nw == 5) {
        // Literal constants -> full unrolling + mul-shift div in staging.
        nnloss_body(5, 5, pred, gt, partials, tile, wsum);
    } else {
        nnloss_body(nh, nw, pred, gt, partials, tile, wsum);
    }
}

__global__ __launch_bounds__(256) void nnloss_reduce(
    const float* __restrict__ partials, int n, float inv_count,
    float* __restrict__ out)
{
    __shared__ float wsum[8];
    const int tid = threadIdx.x;
    float s = 0.0f;
    for (int i = tid; i < n; i += 256) s += partials[i];  // fixed order
#pragma unroll
    for (int off = 16; off > 0; off >>= 1)
        s += __shfl_xor(s, off, 32);
    if ((tid & 31) == 0) wsum[tid >> 5] = s;
    __syncthreads();
    if (tid == 0) {
        float t = 0.0f;
#pragma unroll
        for (int wv = 0; wv < 8; ++wv) t += wsum[wv];
        out[0] = t * inv_count;
    }
}

extern "C" void kernel_launch(void* const* d_in, const int* in_sizes, int n_in,
                              void* d_out, int out_size, void* d_ws, size_t ws_size,
                              hipStream_t stream)
{
    (void)in_sizes; (void)n_in; (void)out_size; (void)ws_size;
    const float* pred = (const float*)d_in[0];
    const float* gt   = (const float*)d_in[1];
    const int*   nh   = (const int*)d_in[2];
    const int*   nw   = (const int*)d_in[3];
    float* partials = (float*)d_ws;           // 512 floats = 2 KB
    float* out      = (float*)d_out;

    dim3 grid(Wn / TLX, Hn / TLY, Bn);        // 4 x 8 x 16 = 512 blocks
    dim3 block(NTX, TLY);                     // 8 x 32 = 256 threads = 8 wave32
    nnloss_min_tile<<<grid, block, 0, stream>>>(pred, gt, nh, nw, partials);

    const int nparts = (Wn / TLX) * (Hn / TLY) * Bn;
    const float inv  = 1.0f / ((float)Bn * (float)Hn * (float)Wn);
    nnloss_reduce<<<1, 256, 0, stream>>>(partials, nparts, inv, out);
}